// MedicalSTSLossFast_85461259256304
// MI455X (gfx1250) — compile-verified
//
#include <hip/hip_runtime.h>

// Problem constants (match reference setup_inputs()).
#define B_N       2
#define S_LEN     2048
#define T_LEN     (S_LEN - 1)        // 2047
#define V_LEN     32000
#define IGNORE_ID (-100)
#define NTHREADS  256
#define F4_PER_ROW   (V_LEN / 4)         // 8000 float4 per row
#define F4_PER_CHUNK (NTHREADS * 4)      // 1024 float4 per block chunk
#define NCHUNK ((F4_PER_ROW + F4_PER_CHUNK - 1) / F4_PER_CHUNK)  // 8

// ---- CDNA5 async global->LDS path (guarded so the file always compiles) ----
#if defined(__gfx1250__) && __has_builtin(__builtin_amdgcn_global_load_async_to_lds_b128)
#define HAVE_ASYNC_LDS 1
typedef int v4i __attribute__((ext_vector_type(4)));
typedef __attribute__((address_space(1))) v4i GlobalV4i;   // prints as "v4i __device__"
typedef __attribute__((address_space(3))) v4i LdsV4i;      // prints as "v4i __shared__"
#else
#define HAVE_ASYNC_LDS 0
#endif

__device__ __forceinline__ void wait_asynccnt_le(int n) {
#if HAVE_ASYNC_LDS
#if __has_builtin(__builtin_amdgcn_s_wait_asynccnt)
    if (n == 0) __builtin_amdgcn_s_wait_asynccnt(0);
    else        __builtin_amdgcn_s_wait_asynccnt(4);
#else
    if (n == 0) asm volatile("s_wait_asynccnt 0" ::: "memory");
    else        asm volatile("s_wait_asynccnt 4" ::: "memory");
#endif
#else
    (void)n;
#endif
}

#if HAVE_ASYNC_LDS
// Each lane issues 4 b128 async copies for chunk c into its own LDS slots.
// Tail addresses are clamped (not predicated) so every wave issues exactly 4
// async instructions per chunk -> ASYNCcnt accounting is uniform.
__device__ __forceinline__ void issue_chunk(const float4* __restrict__ row4,
                                            float4* dst, int c, int tid) {
#pragma unroll
    for (int j = 0; j < 4; ++j) {
        int li = tid + j * NTHREADS;
        int fi = c * F4_PER_CHUNK + li;
        if (fi >= F4_PER_ROW) fi = F4_PER_ROW - 1;  // clamp, garbage ignored on read
        v4i* gp_generic = (v4i*)(row4 + fi);        // reinterpret (drops const), generic AS
        v4i* lp_generic = (v4i*)(dst + li);         // generic AS
        __builtin_amdgcn_global_load_async_to_lds_b128(
            (GlobalV4i*)gp_generic,                 // addrspacecast -> AS1
            (LdsV4i*)lp_generic,                    // addrspacecast -> AS3
            0, 0);
    }
}
#endif

__device__ __forceinline__ void online_update(float4 v, float& m, float& s) {
    float cm = fmaxf(fmaxf(v.x, v.y), fmaxf(v.z, v.w));
    float nm = fmaxf(m, cm);
    s = s * __expf(m - nm) + __expf(v.x - nm) + __expf(v.y - nm) +
        __expf(v.z - nm) + __expf(v.w - nm);
    m = nm;
}

// ---------------------------------------------------------------------------
// Kernel 1: one workgroup per (b, t) row. Online logsumexp over V, then
// ce[r] = max(valid ? lse - x_label : 0, 1e-5).
// ---------------------------------------------------------------------------
__global__ __launch_bounds__(NTHREADS)
void row_lse_kernel(const float* __restrict__ logits,
                    const int* __restrict__ labels,
                    float* __restrict__ ce_out) {
    const int r = blockIdx.x;            // 0 .. B*T-1
    const int b = r / T_LEN;
    const int t = r - b * T_LEN;
    const size_t row_base = ((size_t)b * S_LEN + t) * (size_t)V_LEN;
    const float4* row4 = (const float4*)(logits + row_base);
    const int tid = threadIdx.x;

    float m = -INFINITY, s = 0.0f;

#if HAVE_ASYNC_LDS
    __shared__ float4 stage[2][F4_PER_CHUNK];   // 2 x 16 KB double buffer
    issue_chunk(row4, stage[0], 0, tid);
    for (int c = 0; c < NCHUNK; ++c) {
        if (c + 1 < NCHUNK) {
            issue_chunk(row4, stage[(c + 1) & 1], c + 1, tid);
            wait_asynccnt_le(4);   // current chunk's 4 copies complete
        } else {
            wait_asynccnt_le(0);
        }
        const float4* buf = stage[c & 1];
#pragma unroll
        for (int j = 0; j < 4; ++j) {
            int li = tid + j * NTHREADS;
            int fi = c * F4_PER_CHUNK + li;
            if (fi < F4_PER_ROW) online_update(buf[li], m, s);
        }
    }
#else
    for (int fi = tid; fi < F4_PER_ROW; fi += NTHREADS)
        online_update(row4[fi], m, s);
#endif

    // Block-wide online-softmax combine.
    __shared__ float rm[NTHREADS], rs[NTHREADS];
    rm[tid] = m; rs[tid] = s;
    __syncthreads();
    for (int off = NTHREADS / 2; off > 0; off >>= 1) {
        if (tid < off) {
            float m1 = rm[tid], s1 = rs[tid];
            float m2 = rm[tid + off], s2 = rs[tid + off];
            float nm = fmaxf(m1, m2);
            rs[tid] = s1 * __expf(m1 - nm) + s2 * __expf(m2 - nm);
            rm[tid] = nm;
        }
        __syncthreads();
    }

    if (tid == 0) {
        int  lbl   = labels[(size_t)b * S_LEN + t + 1];
        bool valid = (lbl != IGNORE_ID);
        int  sl    = valid ? lbl : 0;
        float xl   = logits[row_base + (size_t)sl];
        float lse  = rm[0] + __logf(rs[0]);
        float ce   = valid ? (lse - xl) : 0.0f;
        ce_out[r]  = fmaxf(ce, 1e-5f);
    }
}

// ---------------------------------------------------------------------------
// Kernel 2: per-batch weighting + normalization. One block per batch.
// ---------------------------------------------------------------------------
__device__ __forceinline__ void compute_w(int b, int t, float T_ans,
                                          const int* __restrict__ labels,
                                          const float* __restrict__ surp,
                                          float* w, float* vm) {
    int   lbl = labels[(size_t)b * S_LEN + t + 1];
    float v   = (lbl != IGNORE_ID) ? 1.0f : 0.0f;
    float wr  = (lbl >= 1000 && lbl <= 1099) ? 1.0f : 0.5f;       // GAMMA=0.5
    float d   = T_ans - (float)t;
    float wa  = 0.3f + 0.7f * __expf(-(d * d) * (1.0f / 20000.0f)); // ALPHA, 2*SIGMA^2
    float sc  = surp[(size_t)b * S_LEN + t + 1];
    sc = fminf(fmaxf(sc, 1e-5f), 1.0f);
    *w  = wa * wr * sc * v;
    *vm = v;
}

__global__ __launch_bounds__(NTHREADS)
void weight_kernel(const int* __restrict__ labels,
                   const float* __restrict__ surp,
                   const float* __restrict__ ce,
                   float* __restrict__ out_num,
                   float* __restrict__ out_vm) {
    const int b   = blockIdx.x;
    const int tid = threadIdx.x;
    __shared__ int   ired[NTHREADS];
    __shared__ float f1[NTHREADS], f2[NTHREADS];
    __shared__ float s_invden, s_Tans;

    // Pass A: last valid index.
    int lastv = -1;
    for (int t = tid; t < T_LEN; t += NTHREADS) {
        int lbl = labels[(size_t)b * S_LEN + t + 1];
        if (lbl != IGNORE_ID) lastv = t;   // t ascending -> keeps max seen
    }
    ired[tid] = lastv;
    __syncthreads();
    for (int off = NTHREADS / 2; off > 0; off >>= 1) {
        if (tid < off) ired[tid] = max(ired[tid], ired[tid + off]);
        __syncthreads();
    }
    if (tid == 0) {
        int last = ired[0];
        s_Tans = (float)(last >= 0 ? last : T_LEN - 1);
    }
    __syncthreads();
    const float T_ans = s_Tans;

    // Pass B: sum of weights and valid mask.
    float sw = 0.0f, svm = 0.0f;
    for (int t = tid; t < T_LEN; t += NTHREADS) {
        float w, vm;
        compute_w(b, t, T_ans, labels, surp, &w, &vm);
        sw += w; svm += vm;
    }
    f1[tid] = sw; f2[tid] = svm;
    __syncthreads();
    for (int off = NTHREADS / 2; off > 0; off >>= 1) {
        if (tid < off) { f1[tid] += f1[tid + off]; f2[tid] += f2[tid + off]; }
        __syncthreads();
    }
    if (tid == 0) {
        float mean_w = f1[0] / fmaxf(f2[0], 1e-5f);
        s_invden  = 1.0f / fmaxf(mean_w, 1e-8f);
        out_vm[b] = f2[0];
    }
    __syncthreads();
    const float invden = s_invden;

    // Pass C: weighted CE sum.
    float acc = 0.0f;
    for (int t = tid; t < T_LEN; t += NTHREADS) {
        float w, vm;
        compute_w(b, t, T_ans, labels, surp, &w, &vm);
        acc += ce[(size_t)b * T_LEN + t] * w * invden;
    }
    f1[tid] = acc;
    __syncthreads();
    for (int off = NTHREADS / 2; off > 0; off >>= 1) {
        if (tid < off) f1[tid] += f1[tid + off];
        __syncthreads();
    }
    if (tid == 0) out_num[b] = f1[0];
}

// ---------------------------------------------------------------------------
// Kernel 3: final scalar.
// ---------------------------------------------------------------------------
__global__ void finalize_kernel(const float* __restrict__ num,
                                const float* __restrict__ vmv,
                                float* __restrict__ out) {
    float n = 0.0f, v = 0.0f;
    for (int b = 0; b < B_N; ++b) { n += num[b]; v += vmv[b]; }
    float loss = n / fmaxf(v, 1.0f);
    out[0] = (loss != loss) ? 0.0f : loss;
}

extern "C" void kernel_launch(void* const* d_in, const int* in_sizes, int n_in,
                              void* d_out, int out_size, void* d_ws, size_t ws_size,
                              hipStream_t stream) {
    (void)in_sizes; (void)n_in; (void)out_size; (void)ws_size;
    const float* logits = (const float*)d_in[0];   // (B, S, V) f32
    const int*   labels = (const int*)d_in[1];     // (B, S)    i32
    const float* surp   = (const float*)d_in[2];   // (B, S)    f32

    float* ce  = (float*)d_ws;                     // B*T floats
    float* num = ce + (size_t)B_N * T_LEN;         // B floats
    float* vm  = num + B_N;                        // B floats

    row_lse_kernel<<<B_N * T_LEN, NTHREADS, 0, stream>>>(logits, labels, ce);
    weight_kernel<<<B_N, NTHREADS, 0, stream>>>(labels, surp, ce, num, vm);
    finalize_kernel<<<1, 1, 0, stream>>>(num, vm, (float*)d_out);
}